// KMeans_83270825935426
// MI455X (gfx1250) — compile-verified
//
#include <hip/hip_runtime.h>
#include <hip/hip_bf16.h>

#define KN 200000
#define KD 64
#define KK 256

typedef __attribute__((ext_vector_type(2))) float v2f;
typedef __attribute__((ext_vector_type(8))) float v8f;

// ---------------------------------------------------------------------------
// gather initial centroids: cent[k][d] = data[idx[k]][d]
// ---------------------------------------------------------------------------
__global__ void kmeans_gather(const float* __restrict__ data,
                              const int* __restrict__ idx,
                              float* __restrict__ cent) {
    int i = blockIdx.x * blockDim.x + threadIdx.x;     // 0 .. K*D-1
    if (i < KK * KD) {
        int k = i >> 6;         // /KD
        int d = i & 63;
        cent[i] = data[(size_t)idx[k] * KD + d];
    }
}

// ---------------------------------------------------------------------------
// assignment: cid[i] = argmin_k ( -2 * <x_i, c_k> + ||c_k||^2 )
// Block = 256 threads = 8 waves. Each wave computes a 16-row x 256-col tile
// of data @ cent^T using V_WMMA_F32_16X16X4_F32, accumulating over D=64
// in 16 k-steps. Centroids (64KB) + their norms staged in LDS.
// ---------------------------------------------------------------------------
__global__ void kmeans_assign(const float* __restrict__ data,
                              const float* __restrict__ cent,
                              int* __restrict__ cid, int n) {
    extern __shared__ float smem[];
    float* ldsC = smem;            // KK*KD floats (64 KB)
    float* ldsN = smem + KK * KD;  // KK floats

    const int tid = threadIdx.x;

    // stage centroids into LDS (16384 floats / 256 threads = 16 float4 each)
    {
        const float4* src = (const float4*)cent;
        float4* dst = (float4*)ldsC;
        for (int i = tid; i < (KK * KD) / 4; i += 256) dst[i] = src[i];
    }
    __syncthreads();
    // centroid squared norms (one column per thread; K == blockDim)
    {
        float s = 0.0f;
        const float* c = &ldsC[tid * KD];
        #pragma unroll
        for (int d = 0; d < KD; ++d) s = fmaf(c[d], c[d], s);
        ldsN[tid] = s;
    }
    __syncthreads();

    const int wave = tid >> 5;            // 0..7
    const int lane = tid & 31;
    const int half = lane >> 4;           // 0 | 1  (selects K=0,1 vs K=2,3)
    const int l16  = lane & 15;
    const int rowbase = blockIdx.x * 128 + wave * 16;

    // --- preload A tile: 16 k-steps of v2f per lane ----------------------
    // ISA 16x4 f32 A layout: lanes 0-15 -> M=lane, K=0,1 ; lanes 16-31 ->
    // M=lane-16, K=2,3.  So lane loads data[row][4*s + 2*half .. +1].
    int arow = rowbase + l16;
    if (arow >= n) arow = n - 1;          // clamp phantom rows (no store later)
    v2f a[16];
    const float* ap = &data[(size_t)arow * KD + half * 2];
    #pragma unroll
    for (int s = 0; s < 16; ++s) a[s] = *(const v2f*)(ap + 4 * s);

    float minv[8];
    int   mini[8];
    #pragma unroll
    for (int v = 0; v < 8; ++v) { minv[v] = 3.4e38f; mini[v] = 0; }

    // --- 16 column tiles of 16 centroids each ----------------------------
    for (int ct = 0; ct < 16; ++ct) {
        const int col = ct * 16 + l16;
        // B (4x16, KxN) layout mirrors A: lanes 0-15 -> N=lane, K=0,1 ;
        // lanes 16-31 -> N=lane-16, K=2,3.  B[k][n] = cent[col][4*s + k].
        const float* bp = &ldsC[col * KD + half * 2];
        v8f acc = {};
        #pragma unroll
        for (int s = 0; s < 16; ++s) {
            v2f b = *(const v2f*)(bp + 4 * s);
            acc = __builtin_amdgcn_wmma_f32_16x16x4_f32(
                false, a[s], false, b, (short)0, acc, false, false);
        }
        const float cn = ldsN[col];
        #pragma unroll
        for (int v = 0; v < 8; ++v) {
            // d2 (minus the row-constant ||x||^2) = -2*dot + ||c||^2
            float dv = fmaf(acc[v], -2.0f, cn);
            if (dv < minv[v] || (dv == minv[v] && col < mini[v])) {
                minv[v] = dv; mini[v] = col;
            }
        }
    }

    // --- min-reduce across the 16 lanes of each half-wave ----------------
    // (xor offsets < 16 never cross the half boundary in a 32-lane wave)
    #pragma unroll
    for (int off = 1; off < 16; off <<= 1) {
        #pragma unroll
        for (int v = 0; v < 8; ++v) {
            float ov = __shfl_xor(minv[v], off, 32);
            int   oi = __shfl_xor(mini[v], off, 32);
            if (ov < minv[v] || (ov == minv[v] && oi < mini[v])) {
                minv[v] = ov; mini[v] = oi;
            }
        }
    }

    // C/D layout: lanes 0-15 hold rows M=v, lanes 16-31 hold rows M=v+8.
    if (l16 == 0) {
        #pragma unroll
        for (int v = 0; v < 8; ++v) {
            int row = rowbase + half * 8 + v;
            if (row < n) cid[row] = mini[v];
        }
    }
}

// ---------------------------------------------------------------------------
// segment-sum update: LDS-accumulated sums/counts, flushed with global
// float atomics. 64 threads cooperate on one row (thread -> one dim).
// ---------------------------------------------------------------------------
__global__ void kmeans_update(const float* __restrict__ data,
                              const int* __restrict__ cid,
                              float* __restrict__ gsums,
                              float* __restrict__ gcnts, int n) {
    extern __shared__ float smem[];
    float* ls = smem;            // KK*KD
    float* lc = smem + KK * KD;  // KK
    const int tid = threadIdx.x;

    for (int i = tid; i < KK * KD; i += 256) ls[i] = 0.0f;
    if (tid < KK) lc[tid] = 0.0f;
    __syncthreads();

    const int d    = tid & 63;    // dimension owned by this thread
    const int rsub = tid >> 6;    // 0..3 rows in flight per block
    for (int row = blockIdx.x * 4 + rsub; row < n; row += gridDim.x * 4) {
        int c = cid[row];
        atomicAdd(&ls[c * KD + d], data[(size_t)row * KD + d]);  // ds_add_f32
        if (d == 0) atomicAdd(&lc[c], 1.0f);
    }
    __syncthreads();

    for (int i = tid; i < KK * KD; i += 256) atomicAdd(&gsums[i], ls[i]);
    if (tid < KK) atomicAdd(&gcnts[tid], lc[tid]);
}

// ---------------------------------------------------------------------------
// centroids = sums / counts  (0/0 -> NaN, matching the reference)
// ---------------------------------------------------------------------------
__global__ void kmeans_divide(const float* __restrict__ sums,
                              const float* __restrict__ cnts,
                              float* __restrict__ cent) {
    int i = blockIdx.x * blockDim.x + threadIdx.x;
    if (i < KK * KD) cent[i] = sums[i] / cnts[i >> 6];
}

// ---------------------------------------------------------------------------
// one-hot scatter (output pre-zeroed with hipMemsetAsync)
// ---------------------------------------------------------------------------
__global__ void kmeans_onehot(const int* __restrict__ cid,
                              float* __restrict__ out, int n) {
    int i = blockIdx.x * blockDim.x + threadIdx.x;
    if (i < n) out[(size_t)i * KK + cid[i]] = 1.0f;
}

__global__ void kmeans_copy(const float* __restrict__ cent,
                            float* __restrict__ dst) {
    int i = blockIdx.x * blockDim.x + threadIdx.x;
    if (i < KK * KD) dst[i] = cent[i];
}

// ---------------------------------------------------------------------------
extern "C" void kernel_launch(void* const* d_in, const int* in_sizes, int n_in,
                              void* d_out, int out_size, void* d_ws, size_t ws_size,
                              hipStream_t stream) {
    const float* data = (const float*)d_in[0];
    const int*   idx  = (const int*)d_in[1];
    const int    n    = in_sizes[0] / KD;      // 200000
    const int    ITERS = 10;                   // module constant in reference

    float* out = (float*)d_out;

    // workspace layout
    float* cent = (float*)d_ws;                // K*D
    float* sums = cent + KK * KD;              // K*D
    float* cnts = sums + KK * KD;              // K
    int*   cid  = (int*)(cnts + KK);           // N

    const size_t shmem = (size_t)(KK * KD + KK) * sizeof(float);  // 66560 B
    const int    gAssign = (n + 127) / 128;
    const int    gKD     = (KK * KD + 255) / 256;

    // initial centroids + initial assignment
    kmeans_gather<<<gKD, 256, 0, stream>>>(data, idx, cent);
    kmeans_assign<<<gAssign, 256, shmem, stream>>>(data, cent, cid, n);

    for (int it = 0; it < ITERS; ++it) {
        hipMemsetAsync(sums, 0, (size_t)KK * KD * sizeof(float), stream);
        hipMemsetAsync(cnts, 0, (size_t)KK * sizeof(float), stream);
        kmeans_update<<<256, 256, shmem, stream>>>(data, cid, sums, cnts, n);
        kmeans_divide<<<gKD, 256, 0, stream>>>(sums, cnts, cent);
        kmeans_assign<<<gAssign, 256, shmem, stream>>>(data, cent, cid, n);
    }

    // outputs: one-hot labels (N*K) ++ final centroids (K*D)
    hipMemsetAsync(out, 0, (size_t)n * KK * sizeof(float), stream);
    kmeans_onehot<<<(n + 255) / 256, 256, 0, stream>>>(cid, out, n);
    kmeans_copy<<<gKD, 256, 0, stream>>>(cent, out + (size_t)n * KK);
}